// GAT_69501160784057
// MI455X (gfx1250) — compile-verified
//
#include <hip/hip_runtime.h>
#include <hip/hip_bf16.h>

// ---------------------------------------------------------------------------
// 3-layer GAT forward for MI455X (gfx1250), wave32 + WMMA f32 16x16x4.
// ---------------------------------------------------------------------------

#define GAT_N_NODES  50000
#define GAT_N_EDGES  300000
#define GAT_IN_FEATS 128
#define GAT_HIDDEN   64
#define GAT_HEADS    4
#define GAT_CLASSES  47
#define GAT_NEG_SLOPE 0.2f

typedef __attribute__((ext_vector_type(2))) float v2f;
typedef __attribute__((ext_vector_type(8))) float v8f;

// ---------------------------------------------------------------------------
// GEMM: C[M x N] = A[M x K] @ B[K x N], fp32.
// One wave computes a 16(M) x 64(N) strip = 4 WMMA accumulators, so each
// K-step issues 4 v_wmma_f32_16x16x4_f32 reusing a single A fragment.
//
// Out-of-range columns (N=47 layer) are handled by CLAMPING the load address
// (always in-bounds -> no divergent exec-masked loads; WMMA columns are
// independent so the garbage stays in never-stored C columns) and guarding
// only the epilogue store.
//
// Fragment layouts (ISA 7.12.2, wave32):
//   A 16x4 : lane l -> row M=l%16; VGPR0 K=(l/16)*2, VGPR1 K=(l/16)*2+1
//   B 4x16 : VGPR r -> K-row (l/16)*2+r, column N=l%16
//   C 16x16: VGPR r, lanes 0-15 -> M=r, lanes 16-31 -> M=r+8, N=l%16
// ---------------------------------------------------------------------------
template <int K, int N>
__global__ __launch_bounds__(256)
void gat_gemm_wmma_f32(const float* __restrict__ A, const float* __restrict__ B,
                       float* __restrict__ C, int M) {
  constexpr int STRIPS = (N + 63) / 64;
  const int wid  = (int)((blockIdx.x * blockDim.x + threadIdx.x) >> 5);
  const int lane = (int)(threadIdx.x & 31u);
  const int tM = wid / STRIPS;
  const int tS = wid - tM * STRIPS;
  if (tM * 16 >= M) return;                 // wave-uniform exit

  const int idx  = lane & 15;               // row (A) / col-in-tile (B)
  const int half = lane >> 4;               // 0 -> K pair {0,1}, 1 -> {2,3}
  const int arow = tM * 16 + idx;
  const int col0 = tS * 64 + idx;

  // Clamped (always in-bounds) B column addresses: no per-lane load guards.
  int bc0 = col0;      if (bc0 > N - 1) bc0 = N - 1;
  int bc1 = col0 + 16; if (bc1 > N - 1) bc1 = N - 1;
  int bc2 = col0 + 32; if (bc2 > N - 1) bc2 = N - 1;
  int bc3 = col0 + 48; if (bc3 > N - 1) bc3 = N - 1;

  v8f c0 = {0.f,0.f,0.f,0.f,0.f,0.f,0.f,0.f};
  v8f c1 = c0, c2 = c0, c3 = c0;

  const float* __restrict__ Arow = A + (size_t)arow * K;
  for (int k0 = 0; k0 < K; k0 += 4) {
    const int kb = k0 + half * 2;
    const float2 av = *(const float2*)(Arow + kb);   // one 8-byte load
    v2f a; a.x = av.x; a.y = av.y;

    const float* __restrict__ Br0 = B + (size_t)kb * N;
    const float* __restrict__ Br1 = Br0 + N;
    v2f b;
    b.x = Br0[bc0]; b.y = Br1[bc0];
    c0 = __builtin_amdgcn_wmma_f32_16x16x4_f32(false, a, false, b, (short)0, c0, false, false);
    b.x = Br0[bc1]; b.y = Br1[bc1];
    c1 = __builtin_amdgcn_wmma_f32_16x16x4_f32(false, a, false, b, (short)0, c1, false, false);
    b.x = Br0[bc2]; b.y = Br1[bc2];
    c2 = __builtin_amdgcn_wmma_f32_16x16x4_f32(false, a, false, b, (short)0, c2, false, false);
    b.x = Br0[bc3]; b.y = Br1[bc3];
    c3 = __builtin_amdgcn_wmma_f32_16x16x4_f32(false, a, false, b, (short)0, c3, false, false);
  }

  // Epilogue store (only place that needs per-lane column guards).
  const int obase = tM * 16 + half * 8;
  {
    const int col = col0;
    if (col < N) {
#pragma unroll
      for (int r = 0; r < 8; ++r) C[(size_t)(obase + r) * N + col] = c0[r];
    }
  }
  {
    const int col = col0 + 16;
    if (col < N) {
#pragma unroll
      for (int r = 0; r < 8; ++r) C[(size_t)(obase + r) * N + col] = c1[r];
    }
  }
  {
    const int col = col0 + 32;
    if (col < N) {
#pragma unroll
      for (int r = 0; r < 8; ++r) C[(size_t)(obase + r) * N + col] = c2[r];
    }
  }
  {
    const int col = col0 + 48;
    if (col < N) {
#pragma unroll
      for (int r = 0; r < 8; ++r) C[(size_t)(obase + r) * N + col] = c3[r];
    }
  }
}

// ---------------------------------------------------------------------------
// el[n,h] = sum_d feat[n,h,d]*al[h,d] ; er likewise.  H, D compile-time.
// ---------------------------------------------------------------------------
template <int H, int D>
__global__ __launch_bounds__(256)
void gat_compute_lr(const float* __restrict__ feat,
                    const float* __restrict__ al, const float* __restrict__ ar,
                    float* __restrict__ el, float* __restrict__ er,
                    int n_nodes) {
  const int i = (int)(blockIdx.x * blockDim.x + threadIdx.x);
  if (i >= n_nodes * H) return;
  const int h = i % H;
  const float* __restrict__ f  = feat + (size_t)i * D;
  const float* __restrict__ pl = al + (size_t)h * D;
  const float* __restrict__ pr = ar + (size_t)h * D;
  float sl = 0.f, sr = 0.f;
  for (int d = 0; d < D; ++d) {
    const float v = f[d];
    sl += v * pl[d];
    sr += v * pr[d];
  }
  el[i] = sl;
  er[i] = sr;
}

// Order-preserving f32 <-> u32 mapping for deterministic atomic segment-max.
__device__ __forceinline__ unsigned f32_to_ordered(float f) {
  unsigned u = __float_as_uint(f);
  return (u & 0x80000000u) ? ~u : (u | 0x80000000u);
}
__device__ __forceinline__ float ordered_to_f32(unsigned u) {
  unsigned b = (u & 0x80000000u) ? (u & 0x7FFFFFFFu) : ~u;
  return __uint_as_float(b);
}

// e = leaky_relu(el[src]+er[dst]); segment-max over dst via ordered atomicMax.
template <int H>
__global__ __launch_bounds__(256)
void gat_edge_max(const int* __restrict__ src, const int* __restrict__ dst,
                  const float* __restrict__ el, const float* __restrict__ er,
                  float* __restrict__ e_edge, unsigned* __restrict__ emaxb,
                  int E) {
  const int i = (int)(blockIdx.x * blockDim.x + threadIdx.x);
  if (i >= E * H) return;
  const int e = i / H, h = i - e * H;
  const int s = src[e], d = dst[e];
  float x = el[s * H + h] + er[d * H + h];
  x = (x > 0.f) ? x : GAT_NEG_SLOPE * x;
  e_edge[i] = x;
  atomicMax(&emaxb[d * H + h], f32_to_ordered(x));
}

// ee = exp(e - emax[dst]); denom[dst] += ee
template <int H>
__global__ __launch_bounds__(256)
void gat_edge_exp(const int* __restrict__ dst, float* __restrict__ e_edge,
                  const unsigned* __restrict__ emaxb, float* __restrict__ denom,
                  int E) {
  const int i = (int)(blockIdx.x * blockDim.x + threadIdx.x);
  if (i >= E * H) return;
  const int e = i / H, h = i - e * H;
  const int d = dst[e];
  const float em = ordered_to_f32(emaxb[d * H + h]);
  const float ee = __expf(e_edge[i] - em);
  e_edge[i] = ee;
  atomicAdd(&denom[d * H + h], ee);
}

// alpha = ee / max(denom[dst], 1e-9)
template <int H>
__global__ __launch_bounds__(256)
void gat_edge_alpha(const int* __restrict__ dst, float* __restrict__ e_edge,
                    const float* __restrict__ denom, int E) {
  const int i = (int)(blockIdx.x * blockDim.x + threadIdx.x);
  if (i >= E * H) return;
  const int e = i / H, h = i - e * H;
  e_edge[i] = e_edge[i] / fmaxf(denom[dst[e] * H + h], 1e-9f);
}

// out[n, :] starts as bias broadcast (folds the "+ b" epilogue).
template <int HD>
__global__ __launch_bounds__(256)
void gat_init_out_bias(float* __restrict__ out, const float* __restrict__ b,
                       int n_nodes) {
  const int i = (int)(blockIdx.x * blockDim.x + threadIdx.x);
  if (i >= n_nodes * HD) return;
  out[i] = b[i % HD];
}

// out[dst, h, d] += feat[src, h, d] * alpha[e, h]  (thread per edge-feature)
template <int H, int D>
__global__ __launch_bounds__(256)
void gat_edge_aggregate(const int* __restrict__ src, const int* __restrict__ dst,
                        const float* __restrict__ feat,
                        const float* __restrict__ alpha,
                        float* __restrict__ out, int E) {
  constexpr int HD = H * D;
  const long long i = (long long)blockIdx.x * blockDim.x + threadIdx.x;
  if (i >= (long long)E * HD) return;
  const int e = (int)(i / HD);
  const int r = (int)(i - (long long)e * HD);
  const int h = r / D;
  const float a = alpha[e * H + h];
  const int s = src[e], d = dst[e];
  atomicAdd(&out[(size_t)d * HD + r], feat[(size_t)s * HD + r] * a);
}

// ---------------------------------------------------------------------------
// Host-side layer driver (all launches on `stream`, capture-safe).
// ---------------------------------------------------------------------------
namespace {

inline unsigned blocks_for(long long n, int bs) {
  return (unsigned)((n + bs - 1) / bs);
}

struct Scratch {
  float*    feat;    // [N, 256] (layer 3 uses first N*47)
  float*    hA;      // [N, 256]
  float*    hB;      // [N, 256]
  float*    el;      // [N, 4]
  float*    er;      // [N, 4]
  float*    denom;   // [N, 4]
  unsigned* emaxb;   // [N, 4]
  float*    e_edge;  // [E, 4]
};

template <int H, int D, int K>
void run_gat_layer(const float* h_in, const float* W, const float* al,
                   const float* ar, const float* b, float* out,
                   const int* src, const int* dst,
                   const Scratch& s, hipStream_t stream) {
  constexpr int HD = H * D;
  constexpr int STRIPS = (HD + 63) / 64;
  const int N_nodes = GAT_N_NODES;
  const int E = GAT_N_EDGES;

  // 1) feat = h_in @ W   (WMMA f32 16x16x4; one wave per 16x64 strip)
  {
    const int tilesM = (N_nodes + 15) / 16;
    const long long waves = (long long)tilesM * STRIPS;
    gat_gemm_wmma_f32<K, HD><<<blocks_for(waves * 32, 256), 256, 0, stream>>>(
        h_in, W, s.feat, N_nodes);
  }

  // 2) el / er
  gat_compute_lr<H, D><<<blocks_for((long long)N_nodes * H, 256), 256, 0, stream>>>(
      s.feat, al, ar, s.el, s.er, N_nodes);

  // 3) reset segment-max / denom; seed out with bias
  hipMemsetAsync(s.emaxb, 0, (size_t)N_nodes * H * sizeof(unsigned), stream);
  hipMemsetAsync(s.denom, 0, (size_t)N_nodes * H * sizeof(float), stream);
  gat_init_out_bias<HD><<<blocks_for((long long)N_nodes * HD, 256), 256, 0, stream>>>(
      out, b, N_nodes);

  // 4) edge softmax
  gat_edge_max<H><<<blocks_for((long long)E * H, 256), 256, 0, stream>>>(
      src, dst, s.el, s.er, s.e_edge, s.emaxb, E);
  gat_edge_exp<H><<<blocks_for((long long)E * H, 256), 256, 0, stream>>>(
      dst, s.e_edge, s.emaxb, s.denom, E);
  gat_edge_alpha<H><<<blocks_for((long long)E * H, 256), 256, 0, stream>>>(
      dst, s.e_edge, s.denom, E);

  // 5) weighted aggregation into out
  gat_edge_aggregate<H, D><<<blocks_for((long long)E * HD, 256), 256, 0, stream>>>(
      src, dst, s.feat, s.e_edge, out, E);
}

} // namespace

extern "C" void kernel_launch(void* const* d_in, const int* in_sizes, int n_in,
                              void* d_out, int out_size, void* d_ws, size_t ws_size,
                              hipStream_t stream) {
  (void)in_sizes; (void)n_in; (void)out_size; (void)ws_size;

  // setup_inputs() order:
  const float* x   = (const float*)d_in[0];
  const int*   src = (const int*)d_in[1];
  const int*   dst = (const int*)d_in[2];
  const float* W1  = (const float*)d_in[3];
  const float* al1 = (const float*)d_in[4];
  const float* ar1 = (const float*)d_in[5];
  const float* b1  = (const float*)d_in[6];
  const float* W2  = (const float*)d_in[7];
  const float* al2 = (const float*)d_in[8];
  const float* ar2 = (const float*)d_in[9];
  const float* b2  = (const float*)d_in[10];
  const float* W3  = (const float*)d_in[11];
  const float* al3 = (const float*)d_in[12];
  const float* ar3 = (const float*)d_in[13];
  const float* b3  = (const float*)d_in[14];
  float* out = (float*)d_out;

  // Workspace carve-up (floats)
  float* ws = (float*)d_ws;
  size_t o = 0;
  Scratch s;
  s.feat   = ws + o; o += (size_t)GAT_N_NODES * 256;
  s.hA     = ws + o; o += (size_t)GAT_N_NODES * 256;
  s.hB     = ws + o; o += (size_t)GAT_N_NODES * 256;
  s.el     = ws + o; o += (size_t)GAT_N_NODES * GAT_HEADS;
  s.er     = ws + o; o += (size_t)GAT_N_NODES * GAT_HEADS;
  s.denom  = ws + o; o += (size_t)GAT_N_NODES * GAT_HEADS;
  s.emaxb  = (unsigned*)(ws + o); o += (size_t)GAT_N_NODES * GAT_HEADS;
  s.e_edge = ws + o; o += (size_t)GAT_N_EDGES * GAT_HEADS;

  // Layer 1: x[50000,128] -> hA[50000, 4*64]
  run_gat_layer<GAT_HEADS, GAT_HIDDEN, GAT_IN_FEATS>(
      x, W1, al1, ar1, b1, s.hA, src, dst, s, stream);
  // Layer 2: hA -> hB
  run_gat_layer<GAT_HEADS, GAT_HIDDEN, GAT_HEADS * GAT_HIDDEN>(
      s.hA, W2, al2, ar2, b2, s.hB, src, dst, s, stream);
  // Layer 3: hB -> d_out  (H=1, D=47; mean over 1 head is identity)
  run_gat_layer<1, GAT_CLASSES, GAT_HEADS * GAT_HIDDEN>(
      s.hB, W3, al3, ar3, b3, out, src, dst, s, stream);
}